// ACEEmbedAVD_3255585210531
// MI455X (gfx1250) — compile-verified
//
#include <hip/hip_runtime.h>

// ---------------------------------------------------------------------------
// ACE embedding (radial x vector outer products, segment-sum, channel mixing)
// MI455X / gfx1250 plan:
//   * accumulate only 80 symmetric moments per node (8 rad x {1, v, sym(vvT)})
//     -> 16 MB accumulator, fully L2-resident in the 192 MB L2
//   * edge scatter: hardware global_atomic_add_f32 into L2 (unsafeAtomicAdd)
//   * node mixing: fp32 GEMMs (16x8 @ 8xdim) with V_WMMA_F32_16X16X4_F32,
//     LDS layouts transposed so every fragment is one ds_load_b64, outputs
//     stored as contiguous runs so the backend emits wide global stores
// ---------------------------------------------------------------------------

#define NRADC 8
#define MOM   10            // 1 + 3 + 6 (symmetric second moments)
#define APN   (NRADC*MOM)   // 80 floats of accumulated moments per node
#define DIMA  128
#define DIMV  64
#define DIMD  32
#define OUT_PER_NODE (DIMA + DIMV*3 + DIMD*9)   // 608
#define PI_F 3.14159265358979323846f

typedef __attribute__((ext_vector_type(2))) float v2f;
typedef __attribute__((ext_vector_type(8))) float v8f;

// ---------------------------------------------------------------- zero scratch
__global__ void zero_ws(float* __restrict__ A, long n) {
  long i = (long)blockIdx.x * blockDim.x + threadIdx.x;
  long stride = (long)gridDim.x * blockDim.x;
  for (; i < n; i += stride) A[i] = 0.0f;
}

// ------------------------------------------------------------- edge scatter
// rad[c] = cos(pi*c*sqrt(|r|^2/r0)) * relu(1 - |r|^2/r0), r0 = 8
// v      = u / sqrt(1+|u|^2), u = r * (17/8)
// accumulate rad[c] * {1, v, sym(v v^T)}  (80 values) into A[src]
__global__ void edge_scatter(const float* __restrict__ r,
                             const int*   __restrict__ src,
                             float*       __restrict__ A,
                             int nE) {
  int e = blockIdx.x * blockDim.x + threadIdx.x;
  if (e >= nE) return;
  float x = r[3*e+0], y = r[3*e+1], z = r[3*e+2];
  float rsq = fmaf(x, x, fmaf(y, y, z*z));
  float xsq = rsq * 0.125f;            // |r|^2 / r0, r0 = 8
  float env = 1.0f - xsq;
  if (env <= 0.0f) return;             // outside cutoff: all channels zero (~31% of edges)

  // all 8 cos(pi*c*s) from one hardware cos via Chebyshev recurrence
  float s  = sqrtf(xsq);
  float c1 = __cosf(PI_F * s);
  float rad[NRADC];
  rad[0] = env;
  rad[1] = c1 * env;
  float cp = 1.0f, cc = c1, twoc = 2.0f * c1;
#pragma unroll
  for (int c = 2; c < NRADC; ++c) {
    float cn = twoc * cc - cp; cp = cc; cc = cn;
    rad[c] = cn * env;
  }

  const float k = 17.0f / 8.0f;
  float inv = rsqrtf(1.0f + rsq * (k*k));
  float vx = x*k*inv, vy = y*k*inv, vz = z*k*inv;
  float m[MOM] = {1.0f, vx, vy, vz, vx*vx, vy*vy, vz*vz, vx*vy, vx*vz, vy*vz};

  float* base = A + (size_t)src[e] * APN;
#pragma unroll
  for (int c = 0; c < NRADC; ++c) {
    float rc = rad[c];
#pragma unroll
    for (int q = 0; q < MOM; ++q)
      unsafeAtomicAdd(base + c*MOM + q, rc * m[q]);   // global_atomic_add_f32
  }
}

// --------------------------------------------------------------- node mixing
// One 128-thread block = 4 waves; each wave owns a 16-node tile (50000 = 3125*16).
// Fragment layouts per CDNA5 ISA 7.12.2 (V_WMMA_F32_16X16X4_F32):
//   A (16x4 f32, 2 VGPR): lane L<16 -> M=L, K={0,1}; L>=16 -> M=L-16, K={2,3}
//   B (4x16 f32, 2 VGPR): lane L<16 -> N=L, K={0,1}; L>=16 -> N=L-16, K={2,3}
//   C (16x16 f32, 8 VGPR): vgpr q -> M=q (lanes 0-15) / M=q+8 (lanes 16-31)
// LDS is transposed so each fragment pair (k0,k0+1) is contiguous + 8B aligned:
//   sA: [m][node][k]  (per wave, 10*16*8 floats)
//   sW: [col][k]      (per matrix)
__global__ __launch_bounds__(128)
void node_mix(const float* __restrict__ A,
              const float* __restrict__ Wa,
              const float* __restrict__ Wv,
              const float* __restrict__ Wd,
              float* __restrict__ out, int nNodes) {
  __shared__ float sW[NRADC * (DIMA + DIMV + DIMD)];  // 1792 f32 = 7 KB (transposed)
  __shared__ float sA[4 * MOM * 16 * NRADC];          // 5120 f32 = 20 KB (transposed)

  const int tid  = threadIdx.x;
  const int wave = tid >> 5;
  const int lane = tid & 31;

  // stage W transposed: [col*8 + k]
  for (int i = tid; i < NRADC * (DIMA + DIMV + DIMD); i += 128) {
    if (i < NRADC*DIMA) {
      int k = i >> 7, col = i & (DIMA-1);
      sW[col*NRADC + k] = Wa[i];
    } else if (i < NRADC*(DIMA+DIMV)) {
      int j = i - NRADC*DIMA;
      int k = j >> 6, col = j & (DIMV-1);
      sW[NRADC*DIMA + col*NRADC + k] = Wv[j];
    } else {
      int j = i - NRADC*(DIMA+DIMV);
      int k = j >> 5, col = j & (DIMD-1);
      sW[NRADC*(DIMA+DIMV) + col*NRADC + k] = Wd[j];
    }
  }

  const int tile = blockIdx.x * 4 + wave;
  const int nb   = tile * 16;
  const bool active = (nb < nNodes);   // wave-uniform: EXEC all-1 for WMMA
  if (active) {
    // stage + transpose this wave's 16-node moment block: [m][node][k]
    const float* gA = A + (size_t)nb * APN;
    for (int i = lane; i < 16 * APN; i += 32) {
      float val = gA[i];                 // coalesced global read
      int node = i / APN;
      int rem  = i - node * APN;
      int k    = rem / MOM;
      int m    = rem - k * MOM;
      sA[wave*(16*APN) + m*(16*NRADC) + node*NRADC + k] = val;
    }
  }
  __syncthreads();
  if (!active) return;

  const int row  = lane & 15;            // M (A) / N (B) within tile
  const int hk   = (lane >> 4) << 1;     // upper half-wave carries K+2
  const int rofs = (lane >> 4) << 3;     // C fragment row offset (0 or 8)
  const float* tA  = sA + wave * (16*APN);
  const float* tWa = sW;
  const float* tWv = sW + NRADC*DIMA;
  const float* tWd = sW + NRADC*(DIMA+DIMV);

  auto fragA = [&](int m, int kk) -> v2f {       // one ds_load_b64
    const float* p = tA + m*(16*NRADC) + row*NRADC + kk*4 + hk;
    v2f a; a.x = p[0]; a.y = p[1]; return a;
  };
  auto fragB = [&](const float* Wt, int kk) -> v2f {  // one ds_load_b64
    const float* p = Wt + row*NRADC + kk*4 + hk;      // col == row (lane&15)
    v2f b; b.x = p[0]; b.y = p[1]; return b;
  };

  float* outA = out;
  float* outV = out + (size_t)nNodes * DIMA;
  float* outD = out + (size_t)nNodes * (DIMA + DIMV*3);

  // ---- B_a : (16x8) @ (8x128), 8 col-tiles ----
#pragma unroll
  for (int ct = 0; ct < DIMA/16; ++ct) {
    v8f acc = {};
#pragma unroll
    for (int kk = 0; kk < 2; ++kk)
      acc = __builtin_amdgcn_wmma_f32_16x16x4_f32(false, fragA(0, kk),
                                                  false, fragB(tWa + ct*16*NRADC, kk),
                                                  (short)0, acc, false, false);
#pragma unroll
    for (int q = 0; q < 8; ++q)
      outA[(size_t)(nb + q + rofs)*DIMA + ct*16 + row] = acc[q];
  }

  // ---- B_v : 3 components kept live per col-tile -> contiguous 3-float stores ----
#pragma unroll
  for (int ct = 0; ct < DIMV/16; ++ct) {
    v8f acc[3] = {};
#pragma unroll
    for (int t = 0; t < 3; ++t)
#pragma unroll
      for (int kk = 0; kk < 2; ++kk)
        acc[t] = __builtin_amdgcn_wmma_f32_16x16x4_f32(false, fragA(1 + t, kk),
                                                       false, fragB(tWv + ct*16*NRADC, kk),
                                                       (short)0, acc[t], false, false);
#pragma unroll
    for (int q = 0; q < 8; ++q) {
      float* p = outV + (size_t)(nb + q + rofs)*(DIMV*3) + (ct*16 + row)*3;
      p[0] = acc[0][q]; p[1] = acc[1][q]; p[2] = acc[2][q];   // merged wide store
    }
  }

  // ---- B_d : 6 unique symmetric components, fan out to 9 contiguous floats ----
  // moments: m=4..9 = {xx, yy, zz, xy, xz, yz}; (i,j) -> sym slot:
  const int smap[9] = {0,3,4, 3,1,5, 4,5,2};
#pragma unroll
  for (int ct = 0; ct < DIMD/16; ++ct) {
    v8f acc[6] = {};
#pragma unroll
    for (int s = 0; s < 6; ++s)
#pragma unroll
      for (int kk = 0; kk < 2; ++kk)
        acc[s] = __builtin_amdgcn_wmma_f32_16x16x4_f32(false, fragA(4 + s, kk),
                                                       false, fragB(tWd + ct*16*NRADC, kk),
                                                       (short)0, acc[s], false, false);
#pragma unroll
    for (int q = 0; q < 8; ++q) {
      float* p = outD + (size_t)(nb + q + rofs)*(DIMD*9) + (ct*16 + row)*9;
#pragma unroll
      for (int u = 0; u < 9; ++u) p[u] = acc[smap[u]][q];     // 36B contiguous run
    }
  }
}

// ---------------------------------------------------------------------------
extern "C" void kernel_launch(void* const* d_in, const int* in_sizes, int n_in,
                              void* d_out, int out_size, void* d_ws, size_t ws_size,
                              hipStream_t stream) {
  const float* r   = (const float*)d_in[0];
  const int*   src = (const int*)  d_in[1];
  const float* Wa  = (const float*)d_in[2];
  const float* Wv  = (const float*)d_in[3];
  const float* Wd  = (const float*)d_in[4];

  const int nE = in_sizes[0] / 3;
  const int nN = out_size / OUT_PER_NODE;   // 50000

  float* A = (float*)d_ws;                  // nN * 80 f32 = 16 MB (L2-resident)
  const long nAcc = (long)nN * APN;

  zero_ws<<<1024, 256, 0, stream>>>(A, nAcc);
  edge_scatter<<<(nE + 255) / 256, 256, 0, stream>>>(r, src, A, nE);

  const int nTiles = (nN + 15) / 16;        // 3125 (exact)
  node_mix<<<(nTiles + 3) / 4, 128, 0, stream>>>(A, Wa, Wv, Wd, (float*)d_out, nN);
}